// NeuralIDE_42511586296386
// MI455X (gfx1250) — compile-verified
//
#include <hip/hip_runtime.h>
#include <math.h>
#include <stdint.h>

// Problem constants (must match reference)
#define DD   128
#define TT   128
#define NN   32
#define HH   256
#define HKK  64
#define PP   255          // T + (T-1) sample points
#define PN   (PP*NN)      // 8160 rows, divisible by 16

#define KSPLIT   16       // K-chunks for the (256,8192)x(8192,128) GEMM
#define KCHUNK   (HKK*DD/KSPLIT)   // 512

typedef __attribute__((ext_vector_type(2))) float v2f;
typedef __attribute__((ext_vector_type(8))) float v8f;

// ---------------------------------------------------------------------------
// Generic fp32 WMMA GEMM: one wave32 per block computes a 16x16 tile of
//   C = [acc? C : 0] + A(MxK) * B(KxN)  (+bias over columns) (optional tanh)
// using V_WMMA_F32_16X16X4_F32.
// A-frag layout (ISA 7.12.2, 32-bit A 16x4): lane l -> row M = l&15,
//   VGPR0/1 = K = 2*(l>>4) + {0,1}.
// B-frag (4x16): lane l -> col N = l&15, VGPR0/1 = K = 2*(l>>4) + {0,1}.
// C/D (16x16): lane l -> col N = l&15, VGPR v -> row M = v + 8*(l>>4).
// ---------------------------------------------------------------------------
__global__ void gemm_f32_wmma(const float* __restrict__ A, int lda,
                              const float* __restrict__ B, int ldb,
                              float* __restrict__ C, int ldc,
                              int K, const float* __restrict__ bias,
                              int acc, int do_tanh)
{
    const int tm   = blockIdx.x << 4;
    const int tn   = blockIdx.y << 4;
    const int lane = threadIdx.x;        // 0..31, full wave active
    const int half = lane >> 4;
    const int l16  = lane & 15;

    v8f c = {};
    if (acc) {
#pragma unroll
        for (int v = 0; v < 8; ++v)
            c[v] = C[(size_t)(tm + v + 8 * half) * ldc + tn + l16];
    }

    const float* ap = A + (size_t)(tm + l16) * lda + 2 * half;
    const float* bp = B + (size_t)(2 * half) * ldb + tn + l16;
    for (int k0 = 0; k0 < K; k0 += 4) {
        v2f a, b;
        a.x = ap[0];
        a.y = ap[1];
        b.x = bp[0];
        b.y = bp[ldb];
        // 8 args: (neg_a, A, neg_b, B, c_mod, C, reuse_a, reuse_b)
        c = __builtin_amdgcn_wmma_f32_16x16x4_f32(false, a, false, b,
                                                  (short)0, c, false, false);
        ap += 4;
        bp += (size_t)4 * ldb;
    }

#pragma unroll
    for (int v = 0; v < 8; ++v) {
        float val = c[v];
        if (bias) val += bias[tn + l16];
        if (do_tanh) val = tanhf(val);
        C[(size_t)(tm + v + 8 * half) * ldc + tn + l16] = val;
    }
}

// ---------------------------------------------------------------------------
// K-split variant for the deep-K contraction: grid.z = K-chunk index, each
// block writes a partial 16x16 tile into part[z]. 16x more waves in flight
// and a 16x shorter serial WMMA chain -> hides the L2 load latency that the
// 128-wave monolithic version cannot. Partials reduced deterministically.
// ---------------------------------------------------------------------------
__global__ void gemm_f32_wmma_ksplit(const float* __restrict__ A, int lda,
                                     const float* __restrict__ B, int ldb,
                                     float* __restrict__ part, int ldc,
                                     int rowsC, int kchunk)
{
    const int tm   = blockIdx.x << 4;
    const int tn   = blockIdx.y << 4;
    const int kb   = blockIdx.z * kchunk;
    const int lane = threadIdx.x;
    const int half = lane >> 4;
    const int l16  = lane & 15;

    v8f c = {};
    const float* ap = A + (size_t)(tm + l16) * lda + kb + 2 * half;
    const float* bp = B + (size_t)(kb + 2 * half) * ldb + tn + l16;
    for (int k0 = 0; k0 < kchunk; k0 += 4) {
        // pull the B panel 16 K-rows ahead into cache (global_prefetch_b8)
        __builtin_prefetch(bp + (size_t)16 * ldb, 0, 0);
        v2f a, b;
        a.x = ap[0];
        a.y = ap[1];
        b.x = bp[0];
        b.y = bp[ldb];
        c = __builtin_amdgcn_wmma_f32_16x16x4_f32(false, a, false, b,
                                                  (short)0, c, false, false);
        ap += 4;
        bp += (size_t)4 * ldb;
    }

    float* out = part + (size_t)blockIdx.z * rowsC * ldc;
#pragma unroll
    for (int v = 0; v < 8; ++v)
        out[(size_t)(tm + v + 8 * half) * ldc + tn + l16] = c[v];
}

// Deterministic partial reduction: g[idx] += sum_z part[z][idx] (fixed order).
__global__ void k_reduce(const float* __restrict__ part, float* __restrict__ g,
                         int elems, int nchunks)
{
    const int idx = blockIdx.x * blockDim.x + threadIdx.x;
    if (idx >= elems) return;
    float acc = g[idx];
    for (int z = 0; z < nchunks; ++z) acc += part[(size_t)z * elems + idx];
    g[idx] = acc;
}

// ---------------------------------------------------------------------------
// One-time init: broadcast yg = z0; build transposed weights
//   Wp[(c*128+j)*128 + i] = Wk2[c*16384 + i*128 + j]   (4 MB, L2-resident)
//   Bt[j*128 + i]         = bk2[i*128 + j]
// and zero the padding rows (row 255) of S / fbar / g.
// ---------------------------------------------------------------------------
__global__ void k_init(const float* __restrict__ z0, const float* __restrict__ Wk2,
                       const float* __restrict__ bk2, float* __restrict__ yg,
                       float* __restrict__ Wp, float* __restrict__ Bt,
                       float* __restrict__ S, float* __restrict__ fbar,
                       float* __restrict__ g)
{
    const size_t idx    = (size_t)blockIdx.x * blockDim.x + threadIdx.x;
    const size_t stride = (size_t)gridDim.x * blockDim.x;

    if (idx < (size_t)TT * DD) yg[idx] = z0[idx & (DD - 1)];
    if (idx < (size_t)DD * DD) {
        size_t i = idx >> 7, j = idx & 127;
        Bt[j * DD + i] = bk2[i * DD + j];
    }
    for (size_t e = idx; e < (size_t)HKK * DD * DD; e += stride) {
        size_t c = e >> 14;
        size_t rem = e & 16383;
        size_t i = rem >> 7, j = rem & 127;
        Wp[((c << 7) + j) * DD + i] = Wk2[e];
    }
    if (idx < (size_t)HKK * DD) S[(size_t)255 * HKK * DD + idx] = 0.f;
    if (idx < DD) { fbar[255 * DD + idx] = 0.f; g[255 * DD + idx] = 0.f; }
}

// ---------------------------------------------------------------------------
// Natural cubic spline second-derivatives via Thomas algorithm.
// Superdiagonal factors cp[] depend only on t -> computed once into LDS,
// then 128 threads sweep one column of yg each (forward sweep stores into M,
// back-substitution in place). M[0]=M[127]=0 (boundary identity rows).
// ---------------------------------------------------------------------------
__global__ void k_spline(const float* __restrict__ t, const float* __restrict__ yg,
                         float* __restrict__ M)
{
    __shared__ float cp[TT];
    const int d = threadIdx.x;           // 0..127, one spline column each
    if (d == 0) {
        float prev = 0.f;
        for (int i = 1; i <= TT - 2; ++i) {
            float hm = t[i] - t[i - 1], hp = t[i + 1] - t[i];
            float denom = 2.f * (hm + hp) - hm * prev;
            prev = hp / denom;
            cp[i] = prev;
        }
    }
    __syncthreads();

    float dprev = 0.f;
    for (int i = 1; i <= TT - 2; ++i) {
        float hm = t[i] - t[i - 1], hp = t[i + 1] - t[i];
        float sl1 = (yg[(i + 1) * DD + d] - yg[i * DD + d]) / hp;
        float sl0 = (yg[i * DD + d] - yg[(i - 1) * DD + d]) / hm;
        float rhs = 6.f * (sl1 - sl0);
        float cpm = (i == 1) ? 0.f : cp[i - 1];
        float denom = 2.f * (hm + hp) - hm * cpm;
        dprev = (rhs - hm * dprev) / denom;
        M[i * DD + d] = dprev;
    }
    M[0 * DD + d] = 0.f;
    M[(TT - 1) * DD + d] = 0.f;
    float next = 0.f;
    for (int i = TT - 2; i >= 1; --i) {
        float val = M[i * DD + d] - cp[i] * next;
        M[i * DD + d] = val;
        next = val;
    }
}

// ---------------------------------------------------------------------------
// Spline evaluation at s[p,n] = xs[p] * u[n] -> ys[(p*32+n)*128 + d]
// ---------------------------------------------------------------------------
__global__ void k_eval(const float* __restrict__ t, const float* __restrict__ yg,
                       const float* __restrict__ Mc, float* __restrict__ ys)
{
    const int idx = blockIdx.x * blockDim.x + threadIdx.x;
    if (idx >= PN * DD) return;
    const int d  = idx & (DD - 1);
    const int pn = idx >> 7;
    const int n  = pn & (NN - 1);
    const int p  = pn >> 5;

    const float xp = (p < TT) ? t[p] : 0.5f * (t[p - TT] + t[p - TT + 1]);
    const float q  = xp * ((float)n * (1.0f / (NN - 1)));
    float qc = fminf(fmaxf(q, t[0]), t[TT - 1]);

    int i = (int)(qc * (float)(TT - 1));
    i = min(max(i, 0), TT - 2);
    while (i < TT - 2 && qc >= t[i + 1]) ++i;
    while (i > 0 && qc < t[i]) --i;

    const float hi = t[i + 1] - t[i];
    const float a = (t[i + 1] - qc) / hi;
    const float b = (qc - t[i]) / hi;
    const float h26 = hi * hi * (1.0f / 6.0f);
    ys[idx] = a * yg[i * DD + d] + b * yg[(i + 1) * DD + d] +
              ((a * a * a - a) * Mc[i * DD + d] +
               (b * b * b - b) * Mc[(i + 1) * DD + d]) * h26;
}

// ---------------------------------------------------------------------------
// Per-p quadrature contraction (this is what removes the 534 MB K tensor):
//   hden[n,c] = tanh(xs[p]*Wk1[0,c] + s[p,n]*Wk1[1,c] + bk1[c])
//   S[p, c*128+j] = sum_n wq[n]*hden[n,c]*fy[p,n,j]        (64x128 per p)
//   fbar[p,j]     = sum_n wq[n]*fy[p,n,j]
// The 16 KB fy tile is pulled memory->LDS with GLOBAL_LOAD_ASYNC_TO_LDS_B128
// (ASYNCcnt-tracked, no VGPR bounce), overlapped with the tanh() evaluation
// of the weighted hden tile; s_wait_asynccnt 0 + barrier before use.
// ---------------------------------------------------------------------------
__global__ void k_hS(const float* __restrict__ t, const float* __restrict__ fy,
                     const float* __restrict__ Wk1, const float* __restrict__ bk1,
                     float* __restrict__ S, float* __restrict__ fbar)
{
    __shared__ float sfy[NN * DD];
    __shared__ float shw[NN * HKK];
    const int p   = blockIdx.x;          // 0..254
    const int tid = threadIdx.x;         // 256 threads
    const float xp = (p < TT) ? t[p] : 0.5f * (t[p - TT] + t[p - TT + 1]);

    // async copy fy[p] (32x128 f32) into LDS, 16B per issue, 4 per thread
    for (int i4 = tid; i4 < (NN * DD) / 4; i4 += 256) {
        uint32_t lds_off = (uint32_t)(uintptr_t)(&sfy[i4 * 4]); // low 32 bits
        const float* gp = fy + (size_t)p * NN * DD + (size_t)i4 * 4;
        asm volatile("global_load_async_to_lds_b128 %0, %1, off"
                     :: "v"(lds_off), "v"(gp) : "memory");
    }
    // overlapped with the async copy: weighted hden tile (compute-only)
    for (int i = tid; i < NN * HKK; i += 256) {
        int n = i >> 6, c = i & (HKK - 1);
        float s = xp * ((float)n * (1.0f / (NN - 1)));
        float hd = tanhf(xp * Wk1[c] + s * Wk1[HKK + c] + bk1[c]);
        float wqn = (n == 0 || n == NN - 1) ? 0.5f : 1.0f;
        shw[i] = wqn * hd;
    }
    asm volatile("s_wait_asynccnt 0x0" ::: "memory");
    __syncthreads();

    for (int o = tid; o < HKK * DD; o += 256) {
        int c = o >> 7, j = o & (DD - 1);
        float acc = 0.f;
#pragma unroll
        for (int n = 0; n < NN; ++n) acc += shw[n * HKK + c] * sfy[n * DD + j];
        S[(size_t)p * HKK * DD + o] = acc;
    }
    for (int j = tid; j < DD; j += 256) {
        float acc = 0.f;
#pragma unroll
        for (int n = 0; n < NN; ++n) {
            float wqn = (n == 0 || n == NN - 1) ? 0.5f : 1.0f;
            acc += wqn * sfy[n * DD + j];
        }
        fbar[p * DD + j] = acc;
    }
}

// ---------------------------------------------------------------------------
// Scale g by xs[p]/(N-1), composite Simpson + cumulative sum -> new yg.
// Sequential over the 127 steps (inherent recurrence), parallel over D.
// ---------------------------------------------------------------------------
__global__ void k_ode(const float* __restrict__ t, const float* __restrict__ g,
                      const float* __restrict__ z0, float* __restrict__ yg)
{
    const int d = threadIdx.x;           // 0..127
    float acc = z0[d];
    yg[d] = acc;
    for (int k = 0; k < TT - 1; ++k) {
        float x0 = t[k], x1 = t[k + 1];
        float h  = x1 - x0;
        float xm = 0.5f * (x0 + x1);
        float gt0 = g[k * DD + d]          * (x0 * (1.0f / (NN - 1)));
        float gt1 = g[(k + 1) * DD + d]    * (x1 * (1.0f / (NN - 1)));
        float gm  = g[(TT + k) * DD + d]   * (xm * (1.0f / (NN - 1)));
        acc += h * (1.0f / 6.0f) * (gt0 + 4.f * gm + gt1);
        yg[(k + 1) * DD + d] = acc;
    }
}

// ---------------------------------------------------------------------------
extern "C" void kernel_launch(void* const* d_in, const int* in_sizes, int n_in,
                              void* d_out, int out_size, void* d_ws, size_t ws_size,
                              hipStream_t stream)
{
    const float* z0  = (const float*)d_in[0];
    const float* t   = (const float*)d_in[1];
    const float* W1  = (const float*)d_in[2];
    const float* b1  = (const float*)d_in[3];
    const float* W2  = (const float*)d_in[4];
    const float* b2  = (const float*)d_in[5];
    const float* Wk1 = (const float*)d_in[6];
    const float* bk1 = (const float*)d_in[7];
    const float* Wk2 = (const float*)d_in[8];
    const float* bk2 = (const float*)d_in[9];

    float* yg = (float*)d_out;           // (T, 1, D) == T*D, doubles as state
    float* w  = (float*)d_ws;            // ~31.7 MB total
    float* Wp = w;  w += (size_t)HKK * DD * DD;      // 1048576  transposed Wk2
    float* Bt = w;  w += (size_t)DD * DD;            // 16384    transposed bk2
    float* Mc = w;  w += (size_t)TT * DD;            // 16384    spline M
    float* ys = w;  w += (size_t)PN * DD;            // 1044480
    float* h1 = w;  w += (size_t)PN * HH;            // 2088960
    float* fy = w;  w += (size_t)PN * DD;            // 1044480
    float* S  = w;  w += (size_t)256 * HKK * DD;     // 2097152  (row 255 = pad)
    float* fb = w;  w += (size_t)256 * DD;           // 32768    (row 255 = pad)
    float* g  = w;  w += (size_t)256 * DD;           // 32768
    float* pt = w;  w += (size_t)KSPLIT * 256 * DD;  // 524288   K-split partials

    k_init<<<4096, 256, 0, stream>>>(z0, Wk2, bk2, yg, Wp, Bt, S, fb, g);

    for (int it = 0; it < 3; ++it) {
        k_spline<<<1, 128, 0, stream>>>(t, yg, Mc);
        k_eval<<<(PN * DD + 255) / 256, 256, 0, stream>>>(t, yg, Mc, ys);
        // h1 = tanh(ys @ W1 + b1) : (8160,128)@(128,256)
        gemm_f32_wmma<<<dim3(PN / 16, HH / 16), 32, 0, stream>>>(
            ys, DD, W1, HH, h1, HH, DD, b1, 0, 1);
        // fy = h1 @ W2 + b2 : (8160,256)@(256,128)
        gemm_f32_wmma<<<dim3(PN / 16, DD / 16), 32, 0, stream>>>(
            h1, HH, W2, DD, fy, DD, HH, b2, 0, 0);
        k_hS<<<PP, 256, 0, stream>>>(t, fy, Wk1, bk1, S, fb);
        // g = fbar @ Bt : (256,128)@(128,128)   (bias term of K)
        gemm_f32_wmma<<<dim3(256 / 16, DD / 16), 32, 0, stream>>>(
            fb, DD, Bt, DD, g, DD, DD, nullptr, 0, 0);
        // partials[z] = S[:, z*512:(z+1)*512] @ Wp[z*512:(z+1)*512, :]
        gemm_f32_wmma_ksplit<<<dim3(256 / 16, DD / 16, KSPLIT), 32, 0, stream>>>(
            S, HKK * DD, Wp, DD, pt, DD, 256, KCHUNK);
        // g += sum_z partials[z]  (fixed order -> bitwise deterministic)
        k_reduce<<<(256 * DD + 255) / 256, 256, 0, stream>>>(pt, g, 256 * DD, KSPLIT);
        k_ode<<<1, 128, 0, stream>>>(t, g, z0, yg);
    }
}